// maxpooler_ring_79585743994962
// MI455X (gfx1250) — compile-verified
//
#include <hip/hip_runtime.h>

#define NUM_RING 16
#define MAX_RING 1520
#define CNT      1024
#define NPTS     16384
#define CIN      64
#define U_TOT    380      // LTOT / 64
#define LTOT     24320
#define BATCH    16
#define OC       128
#define QSPLIT   2
#define QLEN     32       // 64 / QSPLIT
#define NPART    (3 * QSPLIT * BATCH)   // 96 partial-sum slabs per channel

typedef float v2f __attribute__((ext_vector_type(2)));
typedef float v8f __attribute__((ext_vector_type(8)));
typedef float v4f __attribute__((ext_vector_type(4)));

// monotone float <-> uint mapping so integer atomics implement float max/min
__device__ __forceinline__ unsigned fmap(float f) {
  unsigned u = __float_as_uint(f);
  return (u & 0x80000000u) ? ~u : (u | 0x80000000u);
}
__device__ __forceinline__ float funmap(unsigned m) {
  unsigned u = (m & 0x80000000u) ? (m & 0x7FFFFFFFu) : ~m;
  return __uint_as_float(u);
}

// ---------------- kernel 0: init pooled max/min sentinel arrays ----------------
__global__ void k0_init(unsigned* __restrict__ pmax_u, unsigned* __restrict__ pmin_u, int n) {
  int i = blockIdx.x * blockDim.x + threadIdx.x;
  if (i < n) { pmax_u[i] = 0u; pmin_u[i] = 0xFFFFFFFFu; }
}

// ---------------- kernel 1: fused gather + grouped conv (WMMA) + stats + ring pool ----
// grid: 1536 blocks = (pass 0..2) x (qs 0..1) x (b 0..15) x (g 0..15), 256 threads.
// A wave handles 16 consecutive u; one V_WMMA_F32_16X16X4_F32 per q computes
// y[b, 8g+0..7, (u..u+15)*64+q] with bias folded into the C operand.
// Gathers are software-pipelined one iteration ahead to hide L2 latency.
__global__ __launch_bounds__(256) void k1_conv_stats(
    const float* __restrict__ x, const float* __restrict__ Wg,
    const float* __restrict__ bias, const int* __restrict__ choice,
    unsigned* __restrict__ pmax_u, unsigned* __restrict__ pmin_u,
    float* __restrict__ psum, float* __restrict__ psumsq)
{
  const int bx   = blockIdx.x;
  const int pass = bx >> 9;          // 0..2
  const int qs   = (bx >> 8) & 1;    // q-range half
  const int b    = (bx >> 4) & 15;
  const int g    = bx & 15;
  const int tid  = threadIdx.x;
  const int lane = tid & 31;
  const int wv   = tid >> 5;         // wave 0..7
  const int j    = lane & 15;        // B-matrix column / u offset, A-matrix row
  const int hi   = lane >> 4;        // half-wave: selects K pair {0,1} vs {2,3}

  __shared__ unsigned smax[128];     // [o][r] mapped-uint
  __shared__ unsigned smin[128];
  __shared__ float    sred[256];

  if (tid < 128) { smax[tid] = 0u; smin[tid] = 0xFFFFFFFFu; }
  __syncthreads();

  // A (16x4 f32): lane row M=j, VGPR0 holds K= hi?2:0, VGPR1 holds K= hi?3:1
  v2f A;
  {
    float a0 = 0.f, a1 = 0.f;
    if (j < 8) {
      const float* wp = Wg + (g * 8 + j) * 4;
      a0 = wp[hi ? 2 : 0];
      a1 = wp[hi ? 3 : 1];
    }
    A.x = a0; A.y = a1;
  }
  // C (16x16 f32): VGPR v = row M=v (lanes 0-15). Inject bias; rows 8-15 are don't-care.
  v8f Cb;
#pragma unroll
  for (int v = 0; v < 8; ++v) Cb[v] = bias[g * 8 + v];

  const int  u     = pass * 128 + wv * 16 + j;
  const int  uc    = (u < U_TOT) ? u : (U_TOT - 1);
  const bool valid = (hi == 0) && (u < U_TOT);

  // gather positions for this lane's two B-matrix slots (c = hi?2:0 and c+1)
  const int ca  = hi ? 2 : 0;
  const int ppa = (4 * g + ca) * U_TOT + uc;
  const int ppb = ppa + U_TOT;
  const int* chb = choice + b * (NUM_RING * MAX_RING);
  const int posa = (ppa / MAX_RING) * CNT + chb[ppa];
  const int posb = (ppb / MAX_RING) * CNT + chb[ppb];

  const int q0 = qs * QLEN;
  const float* xb  = x + (size_t)b * (size_t)(CIN * NPTS);
  const float* pxa = xb + posa + (size_t)q0 * NPTS;
  const float* pxb = xb + posb + (size_t)q0 * NPTS;

  float sum[8], sumsq[8], mx[8], mn[8];
#pragma unroll
  for (int v = 0; v < 8; ++v) { sum[v] = 0.f; sumsq[v] = 0.f; mx[v] = -3.4e38f; mn[v] = 3.4e38f; }

  int l    = uc * 64 + q0;
  int ring = l / MAX_RING;
  int bnd  = (ring + 1) * MAX_RING;

  // prologue: load first q of this half
  v2f Bcur;
  Bcur.x = pxa[0];
  Bcur.y = pxb[0];

  for (int qi = 0; qi < QLEN; ++qi) {
    // issue next iteration's gathers before consuming current data
    const int qn = (qi < QLEN - 1) ? (qi + 1) : (QLEN - 1);
    v2f Bnext;
    Bnext.x = pxa[(size_t)qn * NPTS];
    Bnext.y = pxb[(size_t)qn * NPTS];

    // B (4x16 f32): lane column N=j, VGPR0 K= hi?2:0, VGPR1 K= hi?3:1
    v8f D = __builtin_amdgcn_wmma_f32_16x16x4_f32(
        false, A, false, Bcur, (short)0, Cb, false, false);

    // lane crosses at most one ring boundary across its QLEN consecutive l's
    if (valid && l >= bnd) {
#pragma unroll
      for (int v = 0; v < 8; ++v) {
        atomicMax(&smax[v * 16 + ring], fmap(mx[v]));
        atomicMin(&smin[v * 16 + ring], fmap(mn[v]));
        mx[v] = -3.4e38f; mn[v] = 3.4e38f;
      }
      ring++; bnd += MAX_RING;
    }
#pragma unroll
    for (int v = 0; v < 8; ++v) {
      float y  = D[v];
      mx[v]    = fmaxf(mx[v], y);
      mn[v]    = fminf(mn[v], y);
      sum[v]  += y;
      sumsq[v] = fmaf(y, y, sumsq[v]);
    }
    ++l;
    Bcur = Bnext;
  }
  if (valid) {
#pragma unroll
    for (int v = 0; v < 8; ++v) {
      atomicMax(&smax[v * 16 + ring], fmap(mx[v]));
      atomicMin(&smin[v * 16 + ring], fmap(mn[v]));
    }
  }
  __syncthreads();

  // deterministic tree-reduce of sum/sumsq per output channel
  const int slab = (pass * QSPLIT + qs) * BATCH + b;   // 0..95
  for (int v = 0; v < 8; ++v) {
    sred[tid] = valid ? sum[v] : 0.f;
    __syncthreads();
    for (int s = 128; s > 0; s >>= 1) {
      if (tid < s) sred[tid] += sred[tid + s];
      __syncthreads();
    }
    if (tid == 0) psum[slab * OC + g * 8 + v] = sred[0];
    __syncthreads();
    sred[tid] = valid ? sumsq[v] : 0.f;
    __syncthreads();
    for (int s = 128; s > 0; s >>= 1) {
      if (tid < s) sred[tid] += sred[tid + s];
      __syncthreads();
    }
    if (tid == 0) psumsq[slab * OC + g * 8 + v] = sred[0];
    __syncthreads();
  }

  // publish per-(b, oc, ring) max/min via order-independent integer atomics
  if (tid < 128) {
    int o = tid >> 4, r = tid & 15;
    int slot = (b * OC + g * 8 + o) * NUM_RING + r;
    atomicMax(&pmax_u[slot], smax[tid]);
    atomicMin(&pmin_u[slot], smin[tid]);
  }
}

// ---------------- kernel 2: finalize BN affine (s, t) per channel ----------------
__global__ void k2_bnstats(const float* __restrict__ psum, const float* __restrict__ psumsq,
                           const float* __restrict__ gamma, const float* __restrict__ beta,
                           float* __restrict__ sarr, float* __restrict__ tarr)
{
  int oc = threadIdx.x;  // 128 threads
  float s = 0.f, ss = 0.f;
  for (int p = 0; p < NPART; ++p) { s += psum[p * OC + oc]; ss += psumsq[p * OC + oc]; }
  const float inv  = 1.0f / (float)(BATCH * LTOT);
  float mean = s * inv;
  float var  = ss * inv - mean * mean;
  float sc   = gamma[oc] * rsqrtf(var + 1e-5f);
  sarr[oc] = sc;
  tarr[oc] = beta[oc] - sc * mean;
}

// ---------------- kernel 3: broadcast-write output (128 MB, NT store BW) ----------
__global__ __launch_bounds__(256) void k3_out(
    const unsigned* __restrict__ pmax_u, const unsigned* __restrict__ pmin_u,
    const float* __restrict__ sarr, const float* __restrict__ tarr,
    float* __restrict__ out)
{
  const int bo  = blockIdx.x;        // b*128 + oc
  const int oc  = bo & 127;
  const int tid = threadIdx.x;
  __shared__ float rv[16];
  if (tid < 16) {
    float sc = sarr[oc], tt = tarr[oc];
    unsigned um = (sc >= 0.f) ? pmax_u[bo * NUM_RING + tid] : pmin_u[bo * NUM_RING + tid];
    rv[tid] = fmaf(sc, funmap(um), tt);
  }
  __syncthreads();
  v4f* op = (v4f*)out + (size_t)bo * (NPTS / 4);
#pragma unroll
  for (int k = 0; k < 16; ++k) {
    int i4 = tid + k * 256;          // 4096 float4 per (b,oc) row
    float v = rv[i4 >> 8];           // ring = (i4*4) >> 10
    v4f val = {v, v, v, v};
    __builtin_nontemporal_store(val, &op[i4]);
  }
}

extern "C" void kernel_launch(void* const* d_in, const int* in_sizes, int n_in,
                              void* d_out, int out_size, void* d_ws, size_t ws_size,
                              hipStream_t stream)
{
  (void)in_sizes; (void)n_in; (void)out_size; (void)ws_size;
  const float* x      = (const float*)d_in[0];
  const float* Wg     = (const float*)d_in[1];
  const float* bias   = (const float*)d_in[2];
  const float* gamma  = (const float*)d_in[3];
  const float* beta   = (const float*)d_in[4];
  // d_in[5] = ring (unused: structure is implicit)
  const int*   choice = (const int*)d_in[6];

  char* ws = (char*)d_ws;
  unsigned* pmax_u = (unsigned*)ws;                         // 32768 u32
  unsigned* pmin_u = (unsigned*)(ws + 131072);              // 32768 u32
  float*    psum   = (float*)(ws + 262144);                 // 96*128 f32
  float*    psumsq = (float*)(ws + 262144 + 49152);         // 96*128 f32
  float*    sarr   = (float*)(ws + 262144 + 98304);         // 128 f32
  float*    tarr   = (float*)(ws + 262144 + 98304 + 512);   // 128 f32

  const int nslots = BATCH * OC * NUM_RING;                 // 32768
  k0_init<<<(nslots + 255) / 256, 256, 0, stream>>>(pmax_u, pmin_u, nslots);
  k1_conv_stats<<<3 * QSPLIT * BATCH * NUM_RING, 256, 0, stream>>>(
      x, Wg, bias, choice, pmax_u, pmin_u, psum, psumsq);
  k2_bnstats<<<1, 128, 0, stream>>>(psum, psumsq, gamma, beta, sarr, tarr);
  k3_out<<<BATCH * OC, 256, 0, stream>>>(pmax_u, pmin_u, sarr, tarr, (float*)d_out);
}